// Decoder_4698694222138
// MI455X (gfx1250) — compile-verified
//
#include <hip/hip_runtime.h>
#include <hip/hip_bf16.h>
#include <math.h>

// ---------------------------------------------------------------------------
// Fused Bahdanau-attention decoder for MI455X (gfx1250).
//
// Key facts exploited:
//  * Computation is fully independent per batch row b  -> 1 workgroup per row,
//    all 16 scan steps run locally; enc_output[b] is read from HBM exactly once
//    (256 MB total vs ~8.7 GB for the naive schedule).
//  * GRU is called with h0 = zeros in the reference -> new_h = sigmoid(-xz)*tanh(xh),
//    gru_rk (input 9) is mathematically dead.
//  * CDNA5 320KB LDS/WGP holds BOTH enc[b] and enc_w1[b] tiles (2 x 2048x16 f32,
//    rows padded to stride 17 floats for bank-conflict-free column access).
//  * enc_w1 = enc @ w1 computed with V_WMMA_F32_16X16X4_F32 (f32, exact).
// ---------------------------------------------------------------------------

typedef __attribute__((ext_vector_type(2))) float v2f;
typedef __attribute__((ext_vector_type(8))) float v8f;

#define B_    2048
#define T_    2048
#define U_    16
#define O_    8
#define S_    16
#define RS    17          // padded LDS row stride in floats (gcd(17,64)=1)
#define NTH   512
#define NWAVE (NTH / 32)

// LDS layout (float offsets)
#define OFF_ENC    0
#define OFF_ENCW1  (OFF_ENC   + T_ * RS)    // 34816
#define OFF_ATTN   (OFF_ENCW1 + T_ * RS)    // 69632
#define OFF_W1     (OFF_ATTN  + T_)         // 71680
#define OFF_W2     (OFF_W1    + 256)
#define OFF_W2B    (OFF_W2    + 256)
#define OFF_W3     (OFF_W2B   + 16)
#define OFF_GRU    (OFF_W3    + 16)
#define OFF_GRUB   (OFF_GRU   + 768)
#define OFF_DK     (OFF_GRUB  + 48)
#define OFF_DB     (OFF_DK    + 128)
#define OFF_Q      (OFF_DB    + 8)
#define OFF_H      (OFF_Q     + 16)
#define OFF_CTX    (OFF_H     + 16)
#define OFF_RMAX   (OFF_CTX   + 16)
#define OFF_RSUM   (OFF_RMAX  + NWAVE)
#define OFF_SCAL   (OFF_RSUM  + NWAVE)
#define SMEM_FLOATS (OFF_SCAL + 4)          // ~293 KB total, < 320 KB LDS/WGP

__device__ __forceinline__ float fast_tanh(float x) {
#if __has_builtin(__builtin_amdgcn_tanhf)
    return __builtin_amdgcn_tanhf(x);       // gfx1250 v_tanh_f32 (TRANS op)
#else
    return tanhf(x);
#endif
}

__device__ __forceinline__ float wave_reduce_max(float v) {
#pragma unroll
    for (int m = 16; m >= 1; m >>= 1) v = fmaxf(v, __shfl_xor(v, m, 32));
    return v;
}
__device__ __forceinline__ float wave_reduce_sum(float v) {
#pragma unroll
    for (int m = 16; m >= 1; m >>= 1) v += __shfl_xor(v, m, 32);
    return v;
}

__global__ __launch_bounds__(NTH, 1)
void decoder_fused_kernel(const float* __restrict__ enc_output,
                          const float* __restrict__ hidden,
                          const float* __restrict__ w1,
                          const float* __restrict__ w2k,
                          const float* __restrict__ w2b,
                          const float* __restrict__ w3k,
                          const float* __restrict__ w3b,
                          const float* __restrict__ gruk,
                          const float* __restrict__ grub,
                          const float* __restrict__ densek,
                          const float* __restrict__ denseb,
                          float* __restrict__ out)
{
    extern __shared__ float sm[];
    const int tid  = threadIdx.x;
    const int b    = blockIdx.x;
    const int lane = tid & 31;
    const int wv   = tid >> 5;

    float* enc = sm + OFF_ENC;
    float* ew  = sm + OFF_ENCW1;
    float* at  = sm + OFF_ATTN;

    // ---- stage small weights into LDS --------------------------------------
    for (int i = tid; i < 256; i += NTH) { sm[OFF_W1 + i] = w1[i]; sm[OFF_W2 + i] = w2k[i]; }
    for (int i = tid; i < 768; i += NTH) sm[OFF_GRU + i] = gruk[i];
    if (tid < 128) sm[OFF_DK  + tid] = densek[tid];
    if (tid < 48)  sm[OFF_GRUB + tid] = grub[tid];
    if (tid < 16) {
        sm[OFF_W2B + tid] = w2b[tid];
        sm[OFF_W3  + tid] = w3k[tid];
        sm[OFF_H   + tid] = hidden[(size_t)b * U_ + tid];   // scan carry init
    }
    if (tid < 8)   sm[OFF_DB + tid] = denseb[tid];
    if (tid == 0)  sm[OFF_SCAL] = w3b[0];

    // ---- single HBM pass: enc_output[b] (128 KB contiguous) -> LDS ---------
    const float4* src = (const float4*)(enc_output + (size_t)b * T_ * U_);
    for (int i = tid; i < (T_ * U_) / 4; i += NTH) {
        float4 v = src[i];
        int t = i >> 2, c = (i & 3) * 4;
        float* d = enc + t * RS + c;
        d[0] = v.x; d[1] = v.y; d[2] = v.z; d[3] = v.w;
    }
    __syncthreads();

    // ---- WMMA: ew[t,u] = sum_v enc[t,v] * w1[v,u]  (f32, 16x16x4) ----------
    {
        const int m  = lane & 15;        // M row within tile / N col for B & D
        const int kh = lane >> 4;        // half-wave selects K pair {0,1}|{2,3}
        for (int tt = wv; tt < T_ / 16; tt += NWAVE) {
            v8f acc = {};
            const int row = tt * 16 + m;
#pragma unroll
            for (int k0 = 0; k0 < 16; k0 += 4) {
                v2f a, bb;
                // A 16x4 f32 layout: VGPR0 = K even, VGPR1 = K odd per half-wave
                a.x  = enc[row * RS + k0 + kh * 2 + 0];
                a.y  = enc[row * RS + k0 + kh * 2 + 1];
                // B 4x16 f32 layout: row (K) striped across lanes
                bb.x = sm[OFF_W1 + (k0 + kh * 2 + 0) * 16 + m];
                bb.y = sm[OFF_W1 + (k0 + kh * 2 + 1) * 16 + m];
                acc = __builtin_amdgcn_wmma_f32_16x16x4_f32(
                          false, a, false, bb, (short)0, acc, false, false);
            }
            // D layout: VGPR r -> M = r + 8*(lane>=16), N = lane&15
#pragma unroll
            for (int r = 0; r < 8; ++r)
                ew[(tt * 16 + r + kh * 8) * RS + m] = acc[r];
        }
    }
    __syncthreads();

    const float w3b0 = sm[OFF_SCAL];

    // ---- 16 decode steps entirely out of LDS -------------------------------
    for (int s = 0; s < S_; ++s) {
        // q = h @ w2 + w2_b
        if (tid < 16) {
            float a = sm[OFF_W2B + tid];
#pragma unroll
            for (int v = 0; v < 16; ++v) a += sm[OFF_H + v] * sm[OFF_W2 + v * 16 + tid];
            sm[OFF_Q + tid] = a;
        }
        __syncthreads();

        // score[t] = w3_b + sum_u tanh(ew[t,u] + q[u]) * w3[u]   (4 t's/thread)
        float sc[4];
        float lmax = -INFINITY;
#pragma unroll
        for (int i = 0; i < 4; ++i) {
            const int t = tid + i * NTH;
            const float* rowp = ew + t * RS;
            float a = w3b0;
#pragma unroll
            for (int u = 0; u < 16; ++u)
                a += fast_tanh(rowp[u] + sm[OFF_Q + u]) * sm[OFF_W3 + u];
            sc[i] = a;
            lmax  = fmaxf(lmax, a);
        }
        lmax = wave_reduce_max(lmax);
        if (lane == 0) sm[OFF_RMAX + wv] = lmax;
        __syncthreads();
        float mx = sm[OFF_RMAX];
        for (int w = 1; w < NWAVE; ++w) mx = fmaxf(mx, sm[OFF_RMAX + w]);

        // p = exp(score - max); stash in LDS; accumulate sum
        float lsum = 0.f;
#pragma unroll
        for (int i = 0; i < 4; ++i) {
            const int t = tid + i * NTH;
            float p = __expf(sc[i] - mx);
            at[t] = p;
            lsum += p;
        }
        lsum = wave_reduce_sum(lsum);
        if (lane == 0) sm[OFF_RSUM + wv] = lsum;
        if (tid < 16)  sm[OFF_CTX + tid] = 0.f;
        __syncthreads();

        // ctx[u] = sum_t p[t] * enc[t,u]   (thread = (u, 64-wide t chunk))
        {
            const int u = tid & 15, ch = tid >> 4;
            float a = 0.f;
#pragma unroll 8
            for (int j = 0; j < 64; ++j) {
                const int t = ch * 64 + j;
                a += at[t] * enc[t * RS + u];
            }
            atomicAdd(&sm[OFF_CTX + u], a);       // ds_add_f32
        }
        __syncthreads();

        // Degenerate GRU (h0 == 0): new_h = sigmoid(-xz) * tanh(xh)
        if (tid < 16) {
            float sum = sm[OFF_RSUM];
            for (int w = 1; w < NWAVE; ++w) sum += sm[OFF_RSUM + w];
            const float inv = 1.f / sum;
            float xz = sm[OFF_GRUB + tid];
            float xh = sm[OFF_GRUB + 32 + tid];
#pragma unroll
            for (int v = 0; v < 16; ++v) {
                const float c = sm[OFF_CTX + v] * inv;
                xz += c * sm[OFF_GRU + v * 48 + tid];
                xh += c * sm[OFF_GRU + v * 48 + 32 + tid];
            }
            sm[OFF_H + tid] = fast_tanh(xh) / (1.f + __expf(xz));
        }
        __syncthreads();

        // out[b, s, :] = new_h @ dense_k + dense_b
        if (tid < 8) {
            float a = sm[OFF_DB + tid];
#pragma unroll
            for (int u = 0; u < 16; ++u) a += sm[OFF_H + u] * sm[OFF_DK + u * 8 + tid];
            out[(size_t)b * (S_ * O_) + s * O_ + tid] = a;
        }
        // Next iteration's first write (Q) is WAR-safe: last Q read was >=2
        // barriers ago; H was written before the barrier above.
    }
}

extern "C" void kernel_launch(void* const* d_in, const int* in_sizes, int n_in,
                              void* d_out, int out_size, void* d_ws, size_t ws_size,
                              hipStream_t stream) {
    (void)in_sizes; (void)n_in; (void)d_ws; (void)ws_size; (void)out_size;
    // setup_inputs() order:
    // 0 num_inputs(=16, compile-time constant here), 1 enc_output, 2 hidden,
    // 3 w1, 4 w2_k, 5 w2_b, 6 w3_k, 7 w3_b, 8 gru_k, 9 gru_rk (dead: h0==0),
    // 10 gru_b, 11 dense_k, 12 dense_b
    const float* enc = (const float*)d_in[1];
    const float* hid = (const float*)d_in[2];
    const float* w1  = (const float*)d_in[3];
    const float* w2k = (const float*)d_in[4];
    const float* w2b = (const float*)d_in[5];
    const float* w3k = (const float*)d_in[6];
    const float* w3b = (const float*)d_in[7];
    const float* gk  = (const float*)d_in[8];
    const float* gb  = (const float*)d_in[10];
    const float* dk  = (const float*)d_in[11];
    const float* db  = (const float*)d_in[12];
    float* out = (float*)d_out;

    const size_t shmem = (size_t)SMEM_FLOATS * sizeof(float);  // ~293 KB LDS
    decoder_fused_kernel<<<B_, NTH, shmem, stream>>>(
        enc, hid, w1, w2k, w2b, w3k, w3b, gk, gb, dk, db, out);
}